// SparseProjection_26121991094502
// MI455X (gfx1250) — compile-verified
//
#include <hip/hip_runtime.h>

#define HH 120
#define WW 160
#define PP (HH*WW)        // 19200 pixels
#define BB 8
#define CF 32             // image feature channels
#define NI 16             // instances
#define NIF 33            // instance feature channels
#define NREP 7
#define NFEAT 67          // 2 + CF + NIF
#define VOX 0.03f
#define FRDIM 256.0f

typedef __attribute__((ext_vector_type(2))) float v2f;
typedef __attribute__((ext_vector_type(8))) float v8f;

#define AS1 __attribute__((address_space(1)))
#define AS3 __attribute__((address_space(3)))

#if defined(__has_builtin)
#if __has_builtin(__builtin_amdgcn_global_load_async_to_lds_b32) && \
    __has_builtin(__builtin_amdgcn_s_wait_asynccnt)
#define USE_ASYNC_LDS 1
#endif
#endif

// General 4x4 inverse (adjugate). Row-major in/out.
__device__ static void inv4x4(const float* m, float* invOut) {
    float inv[16];
    inv[0]  =  m[5]*m[10]*m[15] - m[5]*m[11]*m[14] - m[9]*m[6]*m[15] + m[9]*m[7]*m[14] + m[13]*m[6]*m[11] - m[13]*m[7]*m[10];
    inv[4]  = -m[4]*m[10]*m[15] + m[4]*m[11]*m[14] + m[8]*m[6]*m[15] - m[8]*m[7]*m[14] - m[12]*m[6]*m[11] + m[12]*m[7]*m[10];
    inv[8]  =  m[4]*m[9]*m[15]  - m[4]*m[11]*m[13] - m[8]*m[5]*m[15] + m[8]*m[7]*m[13] + m[12]*m[5]*m[11] - m[12]*m[7]*m[9];
    inv[12] = -m[4]*m[9]*m[14]  + m[4]*m[10]*m[13] + m[8]*m[5]*m[14] - m[8]*m[6]*m[13] - m[12]*m[5]*m[10] + m[12]*m[6]*m[9];
    inv[1]  = -m[1]*m[10]*m[15] + m[1]*m[11]*m[14] + m[9]*m[2]*m[15] - m[9]*m[3]*m[14] - m[13]*m[2]*m[11] + m[13]*m[3]*m[10];
    inv[5]  =  m[0]*m[10]*m[15] - m[0]*m[11]*m[14] - m[8]*m[2]*m[15] + m[8]*m[3]*m[14] + m[12]*m[2]*m[11] - m[12]*m[3]*m[10];
    inv[9]  = -m[0]*m[9]*m[15]  + m[0]*m[11]*m[13] + m[8]*m[1]*m[15] - m[8]*m[3]*m[13] - m[12]*m[1]*m[11] + m[12]*m[3]*m[9];
    inv[13] =  m[0]*m[9]*m[14]  - m[0]*m[10]*m[13] - m[8]*m[1]*m[14] + m[8]*m[2]*m[13] + m[12]*m[1]*m[10] - m[12]*m[2]*m[9];
    inv[2]  =  m[1]*m[6]*m[15]  - m[1]*m[7]*m[14]  - m[5]*m[2]*m[15] + m[5]*m[3]*m[14] + m[13]*m[2]*m[7]  - m[13]*m[3]*m[6];
    inv[6]  = -m[0]*m[6]*m[15]  + m[0]*m[7]*m[14]  + m[4]*m[2]*m[15] - m[4]*m[3]*m[14] - m[12]*m[2]*m[7]  + m[12]*m[3]*m[6];
    inv[10] =  m[0]*m[5]*m[15]  - m[0]*m[7]*m[13]  - m[4]*m[1]*m[15] + m[4]*m[3]*m[13] + m[12]*m[1]*m[7]  - m[12]*m[3]*m[5];
    inv[14] = -m[0]*m[5]*m[14]  + m[0]*m[6]*m[13]  + m[4]*m[1]*m[14] - m[4]*m[2]*m[13] - m[12]*m[1]*m[6]  + m[12]*m[2]*m[5];
    inv[3]  = -m[1]*m[6]*m[11]  + m[1]*m[7]*m[10]  + m[5]*m[2]*m[11] - m[5]*m[3]*m[10] - m[9]*m[2]*m[7]   + m[9]*m[3]*m[6];
    inv[7]  =  m[0]*m[6]*m[11]  - m[0]*m[7]*m[10]  - m[4]*m[2]*m[11] + m[4]*m[3]*m[10] + m[8]*m[2]*m[7]   - m[8]*m[3]*m[6];
    inv[11] = -m[0]*m[5]*m[11]  + m[0]*m[7]*m[9]   + m[4]*m[1]*m[11] - m[4]*m[3]*m[9]  - m[8]*m[1]*m[7]   + m[8]*m[3]*m[5];
    inv[15] =  m[0]*m[5]*m[10]  - m[0]*m[6]*m[9]   - m[4]*m[1]*m[10] + m[4]*m[2]*m[9]  + m[8]*m[1]*m[6]   - m[8]*m[2]*m[5];
    float det = m[0]*inv[0] + m[1]*inv[4] + m[2]*inv[8] + m[3]*inv[12];
    det = 1.0f / det;
    for (int i = 0; i < 16; ++i) invOut[i] = inv[i] * det;
}

// Per-batch: K_inv, frustum corners -> T = (M @ K_inv)[:3,:] and pad. ws[b*16 + {0..11:T, 12..14:pad}]
__global__ void proj_setup(const float* __restrict__ intr, float* __restrict__ ws) {
    int b = threadIdx.x;
    if (b >= BB) return;
    float K[16], Kinv[16];
    #pragma unroll
    for (int i = 0; i < 16; ++i) K[i] = intr[b*16 + i];
    inv4x4(K, Kinv);

    const float zs[2]  = {0.4f, 6.0f};
    const float pxs[4] = {0.f, (float)WW, (float)WW, 0.f};
    const float pys[4] = {0.f, 0.f, (float)HH, (float)HH};
    float mn[3], mx[3];
    bool first = true;
    for (int zi = 0; zi < 2; ++zi) {
        for (int ci = 0; ci < 4; ++ci) {
            float z = zs[zi];
            float h0 = pxs[ci]*z, h1 = pys[ci]*z, h2 = z, h3 = 1.f;
            for (int r = 0; r < 3; ++r) {
                float v = Kinv[r*4+0]*h0 + Kinv[r*4+1]*h1 + Kinv[r*4+2]*h2 + Kinv[r*4+3]*h3;
                if (first) { mn[r] = v; mx[r] = v; }
                else { mn[r] = fminf(mn[r], v); mx[r] = fmaxf(mx[r], v); }
            }
            first = false;
        }
    }
    float* o = ws + b*16;
    for (int r = 0; r < 3; ++r) {
        float cmin = mn[r] / VOX;
        float cmax = mx[r] / VOX;
        float dims = ceilf(cmax - cmin);
        o[12 + r] = floorf((FRDIM - dims) * 0.5f);          // pad
        for (int c = 0; c < 4; ++c)                          // T[r][c] = Kinv[r][c]/vox - cmin*Kinv[3][c]
            o[r*4 + c] = Kinv[r*4 + c] / VOX - cmin * Kinv[12 + c];
    }
}

// Block = 256 threads = 8 waves; each wave owns a 16-pixel tile.
__global__ void __launch_bounds__(256) proj_main(
    const float* __restrict__ depth,      // (B,1,H,W)
    const float* __restrict__ features,   // (B,CF,H,W)
    const float* __restrict__ masks,      // (B,NI,240,320)
    const int*   __restrict__ locations,  // (B,NI)
    const float* __restrict__ ws,
    float* __restrict__ coords_out,       // (B,P,7,3)
    float* __restrict__ feats_out,        // (B,P,7,67)
    float* __restrict__ valid_out)        // (B,P)
{
    __shared__ float lds_grid[8][16][3];
    __shared__ float lds_frac[8][16];
    __shared__ float lds_img [8][16][CF];   // [m][c] -> store-phase reads conflict-free
    __shared__ float lds_inst[8][16][NIF];

    const int b    = blockIdx.y;
    const int tid  = threadIdx.x;
    const int lane = tid & 31;
    const int wv   = tid >> 5;
    const int base = blockIdx.x * 128 + wv * 16;

    const float* o = ws + b*16;
    float pad3[3];
    pad3[0] = o[12]; pad3[1] = o[13]; pad3[2] = o[14];

    // ---- Kick off image-feature staging global->LDS (ASYNCcnt path on CDNA5)
#ifdef USE_ASYNC_LDS
    for (int idx = lane; idx < 16*CF; idx += 32) {
        int cch = idx >> 4;
        int mm  = idx & 15;
        __builtin_amdgcn_global_load_async_to_lds_b32(
            (AS1 int*)(features + ((size_t)b*CF + cch)*PP + base + mm),
            (AS3 int*)&lds_img[wv][mm][cch],
            0 /*offset*/, 0 /*cpol*/);
    }
#else
    for (int idx = lane; idx < 16*CF; idx += 32) {
        int cch = idx >> 4;
        int mm  = idx & 15;
        lds_img[wv][mm][cch] = features[((size_t)b*CF + cch)*PP + base + mm];
    }
#endif

    const int  m   = lane & 15;          // pixel-in-tile this lane serves in fragments
    const int  p   = base + m;
    const int  py  = p / WW;
    const int  px  = p - py*WW;
    const float z  = depth[(size_t)b*PP + p];
    const bool lo  = (lane < 16);

    // --- A fragment, 16x4 f32: lanes 0-15 hold K=0(v0),K=1(v1); lanes 16-31 hold K=2,K=3.
    //     dp = [x*z, y*z, z, 1]
    v2f a;
    a.x = lo ? (float)px * z : z;
    a.y = lo ? (float)py * z : 1.0f;

    // --- B fragment, 4x16 f32: B[k][n] = T[n][k] for n<3 else 0.
    //     Per-lane load from ws avoids a register-array select tree.
    const int n  = lane & 15;
    const int k0 = lo ? 0 : 1;
    float bx = 0.0f, by = 0.0f;
    if (n < 3) {
        bx = o[n*4 + k0];
        by = o[n*4 + k0 + 2];
    }
    v2f bf = {bx, by};

    // --- D = A x B (+0): D[m][n] = grid coordinate n of pixel m. EXEC is full here.
    v8f cacc = {};
    v8f d = __builtin_amdgcn_wmma_f32_16x16x4_f32(
        /*neg_a=*/false, a, /*neg_b=*/false, bf,
        /*c_mod=*/(short)0, cacc, /*reuse_a=*/false, /*reuse_b=*/false);

    // D layout: VGPR v, lanes 0-15 -> M=v, N=lane; lanes 16-31 -> M=v+8, N=lane-16
    if (n < 3) {
        const int mhi = lo ? 0 : 8;
        #pragma unroll
        for (int v = 0; v < 8; ++v)
            lds_grid[wv][v + mhi][n] = d[v];
    }

    // Zero instance features, then scatter masks (last-writer-wins, sequential j)
    {
        float* fl = &lds_inst[wv][0][0];
        for (int idx = lane; idx < 16*NIF; idx += 32) fl[idx] = 0.0f;
    }
    if (lo) {
        float label = 0.0f;
        #pragma unroll
        for (int j = 0; j < NI; ++j) {
            int   locj = locations[b*NI + j];
            float rmv  = masks[(((size_t)b*NI + j)*240 + 2*py)*320 + 2*px];
            label += rmv;
            if (locj + 1 < NIF)                  // JAX scatter drops OOB (loc==32)
                lds_inst[wv][m][locj + 1] = rmv;
        }
        lds_inst[wv][m][0] = (label == 0.0f) ? 1.0f : 0.0f;

        // frac for the truncated-distance feature (same-wave LDS is in-order)
        float cz = lds_grid[wv][m][2];
        lds_frac[wv][m] = cz - truncf(cz);

        valid_out[(size_t)b*PP + p] = (z != 0.0f) ? 1.0f : 0.0f;
    }

#ifdef USE_ASYNC_LDS
    __builtin_amdgcn_s_wait_asynccnt(0);
#endif
    __syncthreads();

    // --- coords: 16*7*3 = 336 values per wave, coalesced stripe
    for (int idx = lane; idx < 16*NREP*3; idx += 32) {
        int mm  = idx / 21;
        int rem = idx - mm*21;
        int r   = rem / 3;
        int nn  = rem - r*3;
        float v = lds_grid[wv][mm][nn] + pad3[nn] + (nn == 2 ? (float)(r - 3) : 0.0f);
        coords_out[(((size_t)b*PP + base + mm)*NREP + r)*3 + nn] = v;
    }

    // --- feats: each lane owns channels {lane, lane+32, lane+64(<67)}; img/inst values
    //     are r-invariant, so fetch them once per pixel and replicate across the 7 reps.
    {
        const int c1 = lane;           // 0..31 : [0,1]=df, [2..31]=img 0..29
        const int c2 = lane + 32;      // 32..63: [32,33]=img 30,31, [34..63]=inst 0..29
        const int c3 = lane + 64;      // 64..66 valid (lanes 0..2): inst 30..32

        for (int mm = 0; mm < 16; ++mm) {
            float frac = lds_frac[wv][mm];
            float v1 = (c1 >= 2)  ? lds_img[wv][mm][c1 - 2] : 0.0f;
            float v2 = (c2 >= 34) ? lds_inst[wv][mm][c2 - 34]
                                  : lds_img [wv][mm][c2 - 2];
            float v3 = (c3 < NFEAT) ? lds_inst[wv][mm][c3 - 34] : 0.0f;

            float* outp = feats_out + ((size_t)b*PP + base + mm)*NREP*NFEAT;
            #pragma unroll
            for (int r = 0; r < NREP; ++r) {
                float w1 = v1;
                if (c1 < 2) {
                    float df = frac + (float)(r - 3);
                    w1 = (c1 == 0) ? ((df > 0.f) ? 1.f : ((df < 0.f) ? -1.f : 0.f))
                                   : fabsf(df);
                }
                outp[c1] = w1;
                outp[c2] = v2;
                if (c3 < NFEAT) outp[c3] = v3;
                outp += NFEAT;
            }
        }
    }
}

extern "C" void kernel_launch(void* const* d_in, const int* in_sizes, int n_in,
                              void* d_out, int out_size, void* d_ws, size_t ws_size,
                              hipStream_t stream) {
    const float* depth     = (const float*)d_in[0];
    const float* features  = (const float*)d_in[1];
    const float* masks     = (const float*)d_in[2];
    const int*   locations = (const int*)d_in[3];
    const float* intr      = (const float*)d_in[4];
    float* ws = (float*)d_ws;

    float* coords_out = (float*)d_out;
    float* feats_out  = coords_out + (size_t)BB*PP*NREP*3;
    float* valid_out  = feats_out  + (size_t)BB*PP*NREP*NFEAT;

    proj_setup<<<1, 32, 0, stream>>>(intr, ws);
    dim3 grid(PP/128, BB);
    proj_main<<<grid, 256, 0, stream>>>(depth, features, masks, locations, ws,
                                        coords_out, feats_out, valid_out);
}